// MurderTreeV2_32014686224986
// MI455X (gfx1250) — compile-verified
//
#include <hip/hip_runtime.h>

// ---------------------------------------------------------------------------
// MurderTreeV2 for MI455X (gfx1250, wave32).
// All GEMMs are [32 x K] @ [K x N] f32 -> memory-bound weight streaming
// (AI = 16 FLOP/byte, ~1.3 GB of weights/call >> 192MB L2 -> HBM at 23.3TB/s).
// GEMMs use V_WMMA_F32_16X16X4_F32 with a triple-buffered
// GLOBAL_LOAD_ASYNC_TO_LDS_B128 pipeline (ASYNCcnt + s_wait_asynccnt).
// K-tile = 32 rows x 32 cols (5KB); each 128-thread block computes a 32x32
// C tile; steady-state loop has exactly one unconditional wait + barrier +
// issue per tile (last two tiles peeled to keep waits branch-free).
// ---------------------------------------------------------------------------

typedef __attribute__((ext_vector_type(2))) float v2f;
typedef __attribute__((ext_vector_type(8))) float v8f;

#define D_ 768
#define H_ 3072
#define BB 32
#define VOCAB_ 32768

// ---------------- CDNA5 async-copy helpers (inline asm) --------------------
__device__ __forceinline__ void async_copy_b128(const float* gsrc, float* lds_dst) {
  // Generic shared pointer: low 32 bits are the LDS byte address.
  unsigned d = (unsigned)(size_t)lds_dst;
  unsigned long long s = (unsigned long long)(size_t)gsrc;
  asm volatile("global_load_async_to_lds_b128 %0, %1, off"
               :: "v"(d), "v"(s) : "memory");
}
__device__ __forceinline__ void wait_async0() {
  asm volatile("s_wait_asynccnt 0x0" ::: "memory");
}
__device__ __forceinline__ void wait_async2() {
  asm volatile("s_wait_asynccnt 0x2" ::: "memory");
}
__device__ __forceinline__ void wait_async4() {
  asm volatile("s_wait_asynccnt 0x4" ::: "memory");
}

// ---------------- block reduction (blockDim.x == 256) ----------------------
__device__ __forceinline__ float block_reduce_sum(float v, float* red) {
  int tid = threadIdx.x;
  red[tid] = v;
  __syncthreads();
#pragma unroll
  for (int s = 128; s > 0; s >>= 1) {
    if (tid < s) red[tid] += red[tid + s];
    __syncthreads();
  }
  float r = red[0];
  __syncthreads();
  return r;
}

// ---------------- embedding gather + sinusoidal positional mean ------------
__global__ void embed_inject_k(const int* __restrict__ ids,
                               const float* __restrict__ emb,
                               float* __restrict__ inject) {
  int bb = blockIdx.x;            // 32 batches * 3 chunks of 256
  int b = bb / 3;
  int d = (bb % 3) * 256 + threadIdx.x;
  const int half = D_ / 2;
  int dd = (d < half) ? d : d - half;
  float freq = __expf(-(float)dd * (logf(100000.0f) / (float)half));
  float acc = 0.0f, pacc = 0.0f;
  const int* idr = ids + b * 1024;
  for (int t = 0; t < 1024; ++t) {
    int id = idr[t];
    if (t + 1 < 1024)
      __builtin_prefetch(&emb[(long)idr[t + 1] * D_ + d], 0, 0);
    acc += emb[(long)id * D_ + d];
    float ang = (float)t * freq;
    pacc += (d < half) ? sinf(ang) : cosf(ang);
  }
  inject[(long)b * D_ + d] =
      acc * (27.712812921102035f / 1024.0f) + pacc * (1.0f / 1024.0f);
}

// ---------------- h = z_global + 0.3 * masked-mean(z_mem) ------------------
__global__ void mix_k(const float* __restrict__ zg, const float* __restrict__ zm,
                      const unsigned char* __restrict__ mask,
                      float* __restrict__ h) {
  int i = blockIdx.x * 256 + threadIdx.x;   // < 32*768
  float s = 0.0f, wsum = 0.0f;
#pragma unroll
  for (int sl = 0; sl < 12; ++sl) {
    float w = mask[sl] ? 1.0f : 0.0f;
    wsum += w;
    s += w * zm[(long)sl * (BB * D_) + i];
  }
  h[i] = zg[i] + 0.3f * s / fmaxf(wsum, 1.0f);
}

// ---------------- layernorm: out = LN(x)*w + b (+inject) -------------------
__global__ void ln_kernel(const float* __restrict__ x, long x_rs,
                          const float* __restrict__ w, long w_rs,
                          const float* __restrict__ bv, long b_rs,
                          const float* __restrict__ inj,
                          float* __restrict__ out) {
  __shared__ float red[256];
  int rb = blockIdx.x, r = rb >> 5, b = rb & 31;
  int tid = threadIdx.x;
  const float* xr = x + (long)r * x_rs + (long)b * D_;
  float v0 = xr[tid], v1 = xr[tid + 256], v2 = xr[tid + 512];
  float mean = block_reduce_sum(v0 + v1 + v2, red) * (1.0f / D_);
  float d0 = v0 - mean, d1 = v1 - mean, d2 = v2 - mean;
  float var = block_reduce_sum(d0 * d0 + d1 * d1 + d2 * d2, red) * (1.0f / D_);
  float rstd = rsqrtf(var + 1e-5f);
  const float* wr = w + (long)r * w_rs;
  const float* br = bv + (long)r * b_rs;
  float i0 = 0.0f, i1 = 0.0f, i2 = 0.0f;
  if (inj) {
    const float* ir = inj + (long)b * D_;
    i0 = ir[tid]; i1 = ir[tid + 256]; i2 = ir[tid + 512];
  }
  float* o = out + ((long)r * BB + b) * D_;
  o[tid]       = d0 * rstd * wr[tid]       + br[tid]       + i0;
  o[tid + 256] = d1 * rstd * wr[tid + 256] + br[tid + 256] + i1;
  o[tid + 512] = d2 * rstd * wr[tid + 512] + br[tid + 512] + i2;
}

// ---------------- out = maybe_rms( a + gate * c ) --------------------------
__global__ void combine_kernel(const float* __restrict__ a, long a_rs,
                               const float* __restrict__ c, long c_rs,
                               const float* __restrict__ g, long g_rs, long g_bs,
                               int gtanh, int donorm,
                               float* __restrict__ out) {
  __shared__ float red[256];
  int rb = blockIdx.x, r = rb >> 5, b = rb & 31;
  int tid = threadIdx.x;
  const float* ar = a + (long)r * a_rs + (long)b * D_;
  const float* cr = c + (long)r * c_rs + (long)b * D_;
  float y[3];
#pragma unroll
  for (int i = 0; i < 3; ++i) {
    int d = tid + i * 256;
    float gv = 1.0f;
    if (g) {
      gv = g[(long)r * g_rs + (long)b * g_bs + d];
      if (gtanh) gv = tanhf(gv);
    }
    y[i] = ar[d] + gv * cr[d];
  }
  float scale = 1.0f;
  if (donorm) {
    float sq = block_reduce_sum(y[0] * y[0] + y[1] * y[1] + y[2] * y[2], red);
    scale = rsqrtf(sq * (1.0f / D_) + 1e-5f);
  }
  float* o = out + ((long)r * BB + b) * D_;
#pragma unroll
  for (int i = 0; i < 3; ++i) o[tid + i * 256] = y[i] * scale;
}

// ---------------- WMMA f32 GEMM with async-LDS weight pipeline -------------
// A: [R,32,K] (a_rs), W: [R,K,N] row-major (w_rs), C: [R,32,N] (c_rs)
// 128 threads = 4 waves; wave -> (mt in {0,16}, 16-wide n slice).
// K-tile = 32 rows x 32 cols, triple-buffered in LDS (row stride 40 floats
// keeps b128 alignment and puts the two half-waves in disjoint bank ranges).
#define RF 40
#define KT 32
#define TFK (KT * RF)   // floats per LDS tile buffer (1280)

template <int EPI>   // 0 = none, 1 = sigmoid
__global__ void gemm_wmma_k(const float* __restrict__ A, long a_rs,
                            const float* __restrict__ W, long w_rs,
                            float* __restrict__ C, long c_rs,
                            int K, int N) {
  __shared__ __align__(16) float bt[3 * TFK];
  int r = blockIdx.y;
  int tid = threadIdx.x;
  int lane = tid & 31;
  int wave = tid >> 5;
  int mt = (wave & 1) << 4;
  int nw = (wave >> 1) << 4;
  int m = lane & 15;
  int kh = lane >> 4;
  const float* Wb = W + (long)r * w_rs + blockIdx.x * 32;
  // copy assignment: 128 threads x 2 b128 = 32 rows x 32 cols
  int crow = tid >> 3;              // 0..15 (also covers crow+16)
  int cc = (tid & 7) << 2;
  const long rowN = (long)N;
  const float* gnext = Wb + (long)crow * N + cc;
  float* ldst = &bt[crow * RF + cc];
  const long gstep = (long)KT * N;
  int T = K >> 5;                   // K multiple of 32 (768 / 3072)
  int bufn = 0;
  auto issue = [&]() {
    float* l = ldst + bufn * TFK;
    async_copy_b128(gnext, l);
    async_copy_b128(gnext + 16 * rowN, l + 16 * RF);
    gnext += gstep;
    bufn = (bufn == 2) ? 0 : bufn + 1;
  };
  issue();                          // tile 0 -> buf 0
  issue();                          // tile 1 -> buf 1
  const float* ap = A + (long)r * a_rs + (long)(mt + m) * K + 2 * kh;
  const float* lrd = &bt[2 * kh * RF + nw + m];
  v8f acc = {};
  int bufc = 0;
  auto compute = [&]() {
    const float* lp = lrd + bufc * TFK;
    bufc = (bufc == 2) ? 0 : bufc + 1;
#pragma unroll
    for (int u = 0; u < 8; ++u) {
      v2f av = {ap[4 * u], ap[4 * u + 1]};
      v2f bv = {lp[(4 * u) * RF], lp[(4 * u + 1) * RF]};
      acc = __builtin_amdgcn_wmma_f32_16x16x4_f32(
          false, av, false, bv, (short)0, acc, false, false);
    }
    ap += KT;
  };
  // steady state: unconditional wait/barrier/issue/compute per tile
  for (int t = 0; t < T - 2; ++t) {
    wait_async2();     // this wave's tile-t pair done (tile t+1 in flight)
    __syncthreads();   // all waves' tile-t parts visible; buf (t+2)%3 free
    issue();           // tile t+2
    compute();         // tile t
  }
  wait_async2(); __syncthreads(); compute();   // tile T-2
  wait_async0(); __syncthreads(); compute();   // tile T-1
  float* cp = C + (long)r * c_rs + blockIdx.x * 32 + nw + m;
#pragma unroll
  for (int v = 0; v < 8; ++v) {
    int mm = mt + v + (kh << 3);
    float val = acc[v];
    if (EPI == 1) val = 1.0f / (1.0f + __expf(-val));
    cp[(long)mm * N] = val;
  }
}

// ---------------- dual WMMA GEMM with fused SiLU: silu(A@Wg)*(A@Wu) --------
__global__ void dual_gemm_silu_k(const float* __restrict__ A, long a_rs,
                                 const float* __restrict__ Wg,
                                 const float* __restrict__ Wu, long w_rs,
                                 float* __restrict__ C, long c_rs,
                                 int K, int N) {
  __shared__ __align__(16) float btg[3 * TFK];
  __shared__ __align__(16) float btu[3 * TFK];
  int r = blockIdx.y;
  int tid = threadIdx.x;
  int lane = tid & 31;
  int wave = tid >> 5;
  int mt = (wave & 1) << 4;
  int nw = (wave >> 1) << 4;
  int m = lane & 15;
  int kh = lane >> 4;
  int crow = tid >> 3;
  int cc = (tid & 7) << 2;
  const long rowN = (long)N;
  const float* gng = Wg + (long)r * w_rs + blockIdx.x * 32 + (long)crow * N + cc;
  const float* gnu = Wu + (long)r * w_rs + blockIdx.x * 32 + (long)crow * N + cc;
  float* ldstg = &btg[crow * RF + cc];
  float* ldstu = &btu[crow * RF + cc];
  const long gstep = (long)KT * N;
  int T = K >> 5;
  int bufn = 0;
  auto issue = [&]() {
    float* lg = ldstg + bufn * TFK;
    float* lu = ldstu + bufn * TFK;
    async_copy_b128(gng, lg);
    async_copy_b128(gng + 16 * rowN, lg + 16 * RF);
    async_copy_b128(gnu, lu);
    async_copy_b128(gnu + 16 * rowN, lu + 16 * RF);
    gng += gstep;
    gnu += gstep;
    bufn = (bufn == 2) ? 0 : bufn + 1;
  };
  issue();
  issue();
  const float* ap = A + (long)r * a_rs + (long)(mt + m) * K + 2 * kh;
  const float* lrg = &btg[2 * kh * RF + nw + m];
  const float* lru = &btu[2 * kh * RF + nw + m];
  v8f accg = {}, accu = {};
  int bufc = 0;
  auto compute = [&]() {
    const float* lg = lrg + bufc * TFK;
    const float* lu = lru + bufc * TFK;
    bufc = (bufc == 2) ? 0 : bufc + 1;
#pragma unroll
    for (int u = 0; u < 8; ++u) {
      v2f av = {ap[4 * u], ap[4 * u + 1]};
      v2f bg = {lg[(4 * u) * RF], lg[(4 * u + 1) * RF]};
      v2f bu = {lu[(4 * u) * RF], lu[(4 * u + 1) * RF]};
      accg = __builtin_amdgcn_wmma_f32_16x16x4_f32(
          false, av, false, bg, (short)0, accg, false, false);
      accu = __builtin_amdgcn_wmma_f32_16x16x4_f32(
          false, av, false, bu, (short)0, accu, false, false);
    }
    ap += KT;
  };
  for (int t = 0; t < T - 2; ++t) {
    wait_async4();     // tile-t quad done (tile t+1's 4 copies in flight)
    __syncthreads();
    issue();
    compute();
  }
  wait_async4(); __syncthreads(); compute();
  wait_async0(); __syncthreads(); compute();
  float* cp = C + (long)r * c_rs + blockIdx.x * 32 + nw + m;
#pragma unroll
  for (int v = 0; v < 8; ++v) {
    int mm = mt + v + (kh << 3);
    float gg = accg[v], uu = accu[v];
    cp[(long)mm * N] = (gg / (1.0f + __expf(-gg))) * uu;   // silu(g)*u
  }
}

// ---------------- per-batch candidate scoring + top-k mean -----------------
__global__ void topk_mean_k(const float* __restrict__ cands,
                            const float* __restrict__ critic,
                            float* __restrict__ out, int C, int Ksel) {
  __shared__ float red[256];
  __shared__ float sc[16];
  __shared__ int sel[8];
  int b = blockIdx.x, tid = threadIdx.x;
  for (int c = 0; c < C; ++c) {
    float p = 0.0f;
    for (int d = tid; d < D_; d += 256)
      p += cands[((long)c * BB + b) * D_ + d] * critic[d];
    float s = block_reduce_sum(p, red);
    if (tid == 0) sc[c] = s;
  }
  __syncthreads();
  if (tid == 0) {
    unsigned used = 0;
    for (int k = 0; k < Ksel; ++k) {
      int bi = 0;
      float bv = -3.4e38f;
      for (int c = 0; c < C; ++c)
        if (!((used >> c) & 1u) && sc[c] > bv) { bv = sc[c]; bi = c; }
      used |= 1u << bi;
      sel[k] = bi;
    }
  }
  __syncthreads();
  float invk = 1.0f / (float)Ksel;
  for (int d = tid; d < D_; d += 256) {
    float s = 0.0f;
    for (int k = 0; k < Ksel; ++k) s += cands[((long)sel[k] * BB + b) * D_ + d];
    out[(long)b * D_ + d] = s * invk;
  }
}

// ===========================================================================
extern "C" void kernel_launch(void* const* d_in, const int* in_sizes, int n_in,
                              void* d_out, int out_size, void* d_ws, size_t ws_size,
                              hipStream_t stream) {
  (void)in_sizes; (void)n_in; (void)out_size; (void)ws_size;
  // Inputs flattened in setup_inputs() insertion order:
  // 0 input_ids  1 z_global  2 z_mem  3 evolution_mask  4 embed
  // 5..14  tier1.blocks {ln1_w,ln1_b,Wv,Wo,ln2_w,ln2_b,Wg,Wu,Wd,gate}, 15/16 norm_w/b
  // 17..26 tier2.blocks, 27/28 norm_w/b ; 29..38 tier3.blocks, 39/40 norm_w/b
  // 41..50 verifier blocks ; 51..53 up1{Wg,Wu,Wd} ; 54..56 up2{Wg,Wu,Wd}
  // 57 critic ; 58 Wr ; 59 lm_W ; 60 halt_W (unused) ; 61 halt_b (unused)
  const int* ids = (const int*)d_in[0];
  const float* z_global = (const float*)d_in[1];
  const float* z_mem = (const float*)d_in[2];
  const unsigned char* mask = (const unsigned char*)d_in[3];
  const float* embed = (const float*)d_in[4];

  struct Blk {
    const float *ln1w, *ln1b, *Wv, *Wo, *ln2w, *ln2b, *Wg, *Wu, *Wd, *gate;
    int depth;
  };
  auto getblk = [&](int i0, int depth) -> Blk {
    return Blk{(const float*)d_in[i0 + 0], (const float*)d_in[i0 + 1],
               (const float*)d_in[i0 + 2], (const float*)d_in[i0 + 3],
               (const float*)d_in[i0 + 4], (const float*)d_in[i0 + 5],
               (const float*)d_in[i0 + 6], (const float*)d_in[i0 + 7],
               (const float*)d_in[i0 + 8], (const float*)d_in[i0 + 9], depth};
  };
  Blk t1 = getblk(5, 1);
  const float* t1nw = (const float*)d_in[15];
  const float* t1nb = (const float*)d_in[16];
  Blk t2 = getblk(17, 2);
  const float* t2nw = (const float*)d_in[27];
  const float* t2nb = (const float*)d_in[28];
  Blk t3 = getblk(29, 2);
  const float* t3nw = (const float*)d_in[39];
  const float* t3nb = (const float*)d_in[40];
  Blk vb = getblk(41, 4);  // verifier: leading dim 4 == 4 scan steps
  const float* up1Wg = (const float*)d_in[51];
  const float* up1Wu = (const float*)d_in[52];
  const float* up1Wd = (const float*)d_in[53];
  const float* up2Wg = (const float*)d_in[54];
  const float* up2Wu = (const float*)d_in[55];
  const float* up2Wd = (const float*)d_in[56];
  const float* critic = (const float*)d_in[57];
  const float* Wr = (const float*)d_in[58];
  const float* lmW = (const float*)d_in[59];
  float* out = (float*)d_out;

  // ---- workspace layout (f32), ~13.8 MB total ----
  float* p = (float*)d_ws;
  auto alloc = [&](long n) { float* q = p; p += n; return q; };
  const long BD = (long)BB * D_;        // 24576
  const long RBD = 10L * BB * D_;       // 245760 (max 10 candidates)
  float* inject = alloc(BD);
  float* base = alloc(BD);
  float* hn = alloc(RBD);
  float* tt = alloc(RBD);
  float* attn = alloc(RBD);
  float* h2 = alloc(RBD);
  float* h2n = alloc(RBD);
  float* gu = alloc(10L * BB * H_);     // 983040
  float* mres = alloc(RBD);
  float* hA = alloc(RBD);
  float* hB = alloc(RBD);
  float* tout = alloc(RBD);
  float* selm = alloc(BD);
  float* sw = alloc(BD);
  float* tin2 = alloc(BD);
  float* tin3 = alloc(BD);
  float* vA = alloc(BD);
  float* vB = alloc(BD);
  float* gatev = alloc(BD);
  float* newh = alloc(BD);

  // ---- prologue: inject (embedding mean) and memory-mixed base state ----
  embed_inject_k<<<96, 256, 0, stream>>>(ids, embed, inject);
  mix_k<<<96, 256, 0, stream>>>(z_global, z_mem, mask, base);

  // ---- one griffin block step for R parallel candidates at depth `dep` ----
  auto gstep = [&](int R, const Blk& bk, int dep, const float* x, long x_rs,
                   const float* inj, float* outh) {
    long depth = bk.depth;
    long vS = depth * D_;
    long mS = depth * (long)D_ * D_;
    long gS = depth * (long)D_ * H_;
    long dS = depth * (long)H_ * D_;
    ln_kernel<<<R * BB, 256, 0, stream>>>(x, x_rs, bk.ln1w + (long)dep * D_, vS,
                                          bk.ln1b + (long)dep * D_, vS, inj, hn);
    gemm_wmma_k<0><<<dim3(D_ / 32, R), 128, 0, stream>>>(
        hn, 32L * D_, bk.Wv + (long)dep * D_ * D_, mS, tt, 32L * D_, D_, D_);
    gemm_wmma_k<0><<<dim3(D_ / 32, R), 128, 0, stream>>>(
        tt, 32L * D_, bk.Wo + (long)dep * D_ * D_, mS, attn, 32L * D_, D_, D_);
    combine_kernel<<<R * BB, 256, 0, stream>>>(hn, 32L * D_, attn, 32L * D_,
                                               nullptr, 0, 0, 0, 1, h2);
    ln_kernel<<<R * BB, 256, 0, stream>>>(h2, 32L * D_, bk.ln2w + (long)dep * D_,
                                          vS, bk.ln2b + (long)dep * D_, vS,
                                          nullptr, h2n);
    dual_gemm_silu_k<<<dim3(H_ / 32, R), 128, 0, stream>>>(
        h2n, 32L * D_, bk.Wg + (long)dep * D_ * H_, bk.Wu + (long)dep * D_ * H_,
        gS, gu, 32L * H_, D_, H_);
    gemm_wmma_k<0><<<dim3(D_ / 32, R), 128, 0, stream>>>(
        gu, 32L * H_, bk.Wd + (long)dep * H_ * D_, dS, mres, 32L * D_, H_, D_);
    combine_kernel<<<R * BB, 256, 0, stream>>>(h2, 32L * D_, mres, 32L * D_,
                                               bk.gate + (long)dep * D_, vS, 0,
                                               1, 1, outh);
  };

  // ---- tier 1: 10 candidates, depth 1 -> topk 8 -> up1 swiglu + base ----
  gstep(10, t1, 0, base, 0, nullptr, hA);
  ln_kernel<<<10 * BB, 256, 0, stream>>>(hA, 32L * D_, t1nw, 0, t1nb, 0, nullptr, tout);
  topk_mean_k<<<BB, 256, 0, stream>>>(tout, critic, selm, 10, 8);
  dual_gemm_silu_k<<<dim3(H_ / 32, 1), 128, 0, stream>>>(selm, 0, up1Wg, up1Wu, 0,
                                                         gu, 0, D_, H_);
  gemm_wmma_k<0><<<dim3(D_ / 32, 1), 128, 0, stream>>>(gu, 0, up1Wd, 0, sw, 0, H_, D_);
  combine_kernel<<<BB, 256, 0, stream>>>(sw, 0, base, 0, nullptr, 0, 0, 0, 0, tin2);

  // ---- tier 2: 6 candidates, depth 2 -> topk 4 -> up2 swiglu + base ----
  gstep(6, t2, 0, tin2, 0, nullptr, hA);
  gstep(6, t2, 1, hA, 32L * D_, nullptr, hB);
  ln_kernel<<<6 * BB, 256, 0, stream>>>(hB, 32L * D_, t2nw, 0, t2nb, 0, nullptr, tout);
  topk_mean_k<<<BB, 256, 0, stream>>>(tout, critic, selm, 6, 4);
  dual_gemm_silu_k<<<dim3(H_ / 32, 1), 128, 0, stream>>>(selm, 0, up2Wg, up2Wu, 0,
                                                         gu, 0, D_, H_);
  gemm_wmma_k<0><<<dim3(D_ / 32, 1), 128, 0, stream>>>(gu, 0, up2Wd, 0, sw, 0, H_, D_);
  combine_kernel<<<BB, 256, 0, stream>>>(sw, 0, base, 0, nullptr, 0, 0, 0, 0, tin3);

  // ---- tier 3: 4 candidates, depth 2 -> argmax (topk with k=1) ----
  gstep(4, t3, 0, tin3, 0, nullptr, hA);
  gstep(4, t3, 1, hA, 32L * D_, nullptr, hB);
  ln_kernel<<<4 * BB, 256, 0, stream>>>(hB, 32L * D_, t3nw, 0, t3nb, 0, nullptr, tout);
  topk_mean_k<<<BB, 256, 0, stream>>>(tout, critic, vA, 4, 1);

  // ---- verifier: 4 sequential griffin steps with inject ----
  float* cur = vA;
  float* nxt = vB;
  for (int s = 0; s < 4; ++s) {
    gstep(1, vb, s, cur, 0, inject, nxt);
    float* tmp = cur; cur = nxt; nxt = tmp;
  }
  // result in `cur`

  // ---- gate = sigmoid(base @ Wr); new_h = rms(base + gate * verify) ----
  gemm_wmma_k<1><<<dim3(D_ / 32, 1), 128, 0, stream>>>(base, 0, Wr, 0, gatev, 0, D_, D_);
  combine_kernel<<<BB, 256, 0, stream>>>(base, 0, cur, 0, gatev, 0, D_, 0, 1, newh);

  // ---- logits = new_h @ lm_W  ([32,768] @ [768,32768] -> d_out) ----
  gemm_wmma_k<0><<<dim3(VOCAB_ / 32, 1), 128, 0, stream>>>(newh, 0, lmW, 0, out, 0,
                                                           D_, VOCAB_);
}